// Multi_Threshold_Acc_77790447665547
// MI455X (gfx1250) — compile-verified
//
#include <hip/hip_runtime.h>

// Multi-threshold LIF (T=4 scan, 4 threshold blocks) — pure streaming kernel.
// Memory-bound: 154 MB in + 154 MB out => ~13 us floor at 23.3 TB/s.
// Each thread owns 4 consecutive channel elements (float4 / b128 traffic),
// issues all 4 timestep loads up front for MLP, keeps the 16 membrane
// potentials in VGPRs, and streams 4 b128 stores with non-temporal hints.

namespace {
constexpr int  kT = 4;                       // timesteps
constexpr int  kJ = 4;                       // threshold blocks
constexpr long kM = 128L * 196L * 384L;      // elements per timestep = 9,633,792
typedef float f4 __attribute__((ext_vector_type(4)));
}

__global__ __launch_bounds__(256) void lif_multi_threshold_kernel(
    const float* __restrict__ x,      // [T, kM]
    const float* __restrict__ th,     // [4]
    float* __restrict__ out) {        // [T, kM]
  const long i    = (long)blockIdx.x * blockDim.x + threadIdx.x;
  const long base = i * 4;
  if (base >= kM) return;

  // Uniform thresholds -> scalar loads (SGPRs).
  const float thv[kJ] = {th[0], th[1], th[2], th[3]};

  // Load all T timesteps up front: 4 outstanding global_load_b128 per thread.
  f4 xt[kT];
#pragma unroll
  for (int t = 0; t < kT; ++t) {
    xt[t] = __builtin_nontemporal_load(
        reinterpret_cast<const f4*>(x + (long)t * kM + base));
  }

  // Membrane potentials: kJ blocks x 4 vector lanes, all in registers.
  float v[kJ][4];
#pragma unroll
  for (int j = 0; j < kJ; ++j)
#pragma unroll
    for (int k = 0; k < 4; ++k) v[j][k] = 0.0f;

#pragma unroll
  for (int t = 0; t < kT; ++t) {
    // Charge uses tau=2: v' = 0.5*v + 0.5*x. Precompute 0.5*x once, shared
    // across all 4 threshold blocks (1 mul amortized over 4 FMAs).
    float hx[4];
#pragma unroll
    for (int k = 0; k < 4; ++k) hx[k] = 0.5f * xt[t][k];

    f4 fire_cnt = {0.0f, 0.0f, 0.0f, 0.0f};
#pragma unroll
    for (int j = 0; j < kJ; ++j) {
#pragma unroll
      for (int k = 0; k < 4; ++k) {
        float vv = __builtin_fmaf(v[j][k], 0.5f, hx[k]);  // charge
        const bool fire = (vv >= thv[j]);                 // Heaviside(v - th)
        fire_cnt[k] += fire ? 1.0f : 0.0f;                // accumulate spikes
        v[j][k] = fire ? 0.0f : vv;                       // hard reset to 0
      }
    }

    __builtin_nontemporal_store(
        fire_cnt, reinterpret_cast<f4*>(out + (long)t * kM + base));
  }
}

extern "C" void kernel_launch(void* const* d_in, const int* in_sizes, int n_in,
                              void* d_out, int out_size, void* d_ws, size_t ws_size,
                              hipStream_t stream) {
  const float* x  = (const float*)d_in[0];   // [4,128,196,384] fp32
  const float* th = (const float*)d_in[1];   // [4] fp32
  float* out = (float*)d_out;                // [4,128,196,384] fp32

  const long threads = kM / 4;                       // 2,408,448 (exact)
  const int  block   = 256;                          // 8 wave32 waves
  const long grid    = (threads + block - 1) / block; // 9408 (exact, no tail)

  lif_multi_threshold_kernel<<<dim3((unsigned)grid), dim3(block), 0, stream>>>(
      x, th, out);
}